// SelectToken_35656818491877
// MI455X (gfx1250) — compile-verified
//
#include <hip/hip_runtime.h>
#include <math.h>

typedef __attribute__((ext_vector_type(2))) float v2f;
typedef __attribute__((ext_vector_type(8))) float v8f;

#define B_    64
#define NT    64
#define NS    4096
#define C_    256
#define HS    64       // 64x64 token image
#define NWIN  1024     // 32x32 windows of 2x2
#define TOPK_ 3
#define PI_F  3.14159265358979323846f

// workspace layout (units: floats)
#define WS_ZMAX 0                        // B_*C_    = 16384
#define WS_WSUM (WS_ZMAX + B_ * C_)      // B_*NWIN  = 65536
#define WS_IDX  (WS_WSUM + B_ * NWIN)    // B_*TOPK_ ints
#define WS_GMAP (WS_IDX + B_ * TOPK_)    // 16 floats

// ---------------------------------------------------------------------------
// 1) z_max[b,c] = max over Nt tokens of z[b,n,c]
// ---------------------------------------------------------------------------
__global__ void zmax_kernel(const float* __restrict__ z, float* __restrict__ zmax) {
  int b = blockIdx.x;
  int c = threadIdx.x;
  const float* p = z + (size_t)b * NT * C_ + c;
  float m = p[0];
  #pragma unroll 4
  for (int n = 1; n < NT; ++n) m = fmaxf(m, p[(size_t)n * C_]);
  zmax[b * C_ + c] = m;
}

// ---------------------------------------------------------------------------
// 2) gmap1[p] = 1 + sum_k w[k] * gabor_k(4x4 grid at pixel p)
// ---------------------------------------------------------------------------
__global__ void gmap_kernel(const float* __restrict__ th_, const float* __restrict__ sg_,
                            const float* __restrict__ lm_, const float* __restrict__ ps_,
                            const float* __restrict__ gm_, const float* __restrict__ wt_,
                            float* __restrict__ gmap1) {
  int p = threadIdx.x;              // 0..15
  int gy = p >> 2, gx = p & 3;
  float yy = (float)gy - 1.5f;
  float xx = (float)gx - 1.5f;
  float acc = 0.f;
  for (int k = 0; k < 16; ++k) {
    float th = th_[k] * (2.f * PI_F);
    float s = sinf(th), c = cosf(th);
    float xp = xx * c + yy * s;
    float yp = -xx * s + yy * c;
    float sig = sg_[k] + 0.5f;
    float lmb = lm_[k] + 0.5f;
    float gam = gm_[k];
    float env = expf(-(xp * xp + gam * gam * yp * yp) / (2.f * sig * sig));
    float car = cosf(2.f * PI_F * xp / lmb + ps_[k]);
    acc += wt_[k] * env * car;
  }
  gmap1[p] = acc + 1.f;
}

// ---------------------------------------------------------------------------
// 3) Streaming pass over x: per block = (batch, group of 16 windows in one
//    window-row). 256 threads accumulate the 4-token channel sums S[16][256]
//    in LDS, then wave 0 computes win_sum[w] = dot(S[w,:], z_max[b,:]) with
//    V_WMMA_F32_16X16X4_F32 (M=16 windows, K=256 in steps of 4, B broadcast).
// ---------------------------------------------------------------------------
__global__ void winsum_kernel(const float* __restrict__ x,
                              const float* __restrict__ zmax,
                              float* __restrict__ wsum) {
  __shared__ float S[16 * 260];     // padded pitch 260: stride%64 == 4 -> no bank conflicts
  __shared__ float zs[C_];
  int b   = blockIdx.x >> 6;
  int g16 = blockIdx.x & 63;        // 64 groups of 16 windows per batch
  int wr  = g16 >> 1;               // window row (0..31)
  int tc0 = (g16 & 1) << 5;         // token-column base (0 or 32)
  int tid = threadIdx.x;
  int lw  = tid >> 4;               // window within group (0..15)
  int cg  = tid & 15;               // 16-channel chunk   (0..15)

  zs[tid] = zmax[b * C_ + tid];

  float4 acc[4] = {};
  const float* xb = x + (size_t)b * NS * C_;
  #pragma unroll
  for (int iy = 0; iy < 2; ++iy) {
    #pragma unroll
    for (int ix = 0; ix < 2; ++ix) {
      int n = (wr * 2 + iy) * HS + tc0 + lw * 2 + ix;
      const float4* r = (const float4*)(xb + (size_t)n * C_ + cg * 16);
      #pragma unroll
      for (int j = 0; j < 4; ++j) {
        float4 v = r[j];
        acc[j].x += v.x; acc[j].y += v.y; acc[j].z += v.z; acc[j].w += v.w;
      }
    }
  }
  float* srow = &S[lw * 260 + cg * 16];
  #pragma unroll
  for (int j = 0; j < 4; ++j) {
    srow[j * 4 + 0] = acc[j].x; srow[j * 4 + 1] = acc[j].y;
    srow[j * 4 + 2] = acc[j].z; srow[j * 4 + 3] = acc[j].w;
  }
  __syncthreads();

  if (tid < 32) {                   // wave 0, EXEC all ones for WMMA
    int lane = tid;
    int m    = lane & 15;           // matrix row (window)
    int koff = (lane >> 4) << 1;    // lanes 16-31 hold K=2,3
    const float* arow = &S[m * 260];
    v8f d = {};
    for (int k0 = 0; k0 < C_; k0 += 4) {
      v2f a, bv;
      a.x  = arow[k0 + koff];
      a.y  = arow[k0 + koff + 1];
      bv.x = zs[k0 + koff];         // B[k,n] = zmax[k] for all n
      bv.y = zs[k0 + koff + 1];
      d = __builtin_amdgcn_wmma_f32_16x16x4_f32(false, a, false, bv,
                                                (short)0, d, false, false);
    }
    // D column 0: lane 0 -> rows 0..7 in d[0..7]; lane 16 -> rows 8..15
    if (m == 0) {
      int rbase = (lane >> 4) * 8;
      float* o = wsum + b * NWIN + g16 * 16;
      #pragma unroll
      for (int r = 0; r < 8; ++r) o[rbase + r] = d[r];
    }
  }
}

// ---------------------------------------------------------------------------
// 4) top-3 windows per batch (strict '>' => lowest index wins ties, matching
//    jax.lax.top_k). win_mean = win_sum/4 is order-equivalent to win_sum.
// ---------------------------------------------------------------------------
__global__ void topk_kernel(const float* __restrict__ wsum, int* __restrict__ idx) {
  int b = blockIdx.x * blockDim.x + threadIdx.x;
  if (b >= B_) return;
  const float* v = wsum + b * NWIN;
  int i0 = -1, i1 = -1;
  for (int t = 0; t < TOPK_; ++t) {
    float best = -INFINITY;
    int bi = 0;
    for (int w = 0; w < NWIN; ++w) {
      if (w == i0 || w == i1) continue;
      float val = v[w];
      if (val > best) { best = val; bi = w; }
    }
    idx[b * TOPK_ + t] = bi;
    if (t == 0) i0 = bi; else i1 = bi;
  }
}

// ---------------------------------------------------------------------------
// 5) Gather selected 2x2 windows, bilinear 2x2 -> 4x4 (jax half-pixel rule),
//    scale by (gmap+1), write (B, 48, C).
// ---------------------------------------------------------------------------
__global__ void gather_kernel(const float* __restrict__ x, const int* __restrict__ idx,
                              const float* __restrict__ gmap1, float* __restrict__ out) {
  int bt = blockIdx.x;              // b*TOPK_ + t
  int b  = bt / TOPK_;
  int c  = threadIdx.x;
  __shared__ float gm[16];
  __shared__ int wsh;
  if (c < 16) gm[c] = gmap1[c];
  if (c == 0) wsh = idx[bt];
  __syncthreads();
  int w  = wsh;
  int wr = w >> 5, wc = w & 31;

  const float* xb = x + (size_t)b * NS * C_ + c;
  float xv[2][2];
  #pragma unroll
  for (int iy = 0; iy < 2; ++iy)
    #pragma unroll
    for (int ix = 0; ix < 2; ++ix) {
      int n = (wr * 2 + iy) * HS + wc * 2 + ix;
      xv[iy][ix] = xb[(size_t)n * C_];
    }

  const float W[4][2] = {{1.f, 0.f}, {0.75f, 0.25f}, {0.25f, 0.75f}, {0.f, 1.f}};
  float* ob = out + (size_t)bt * 16 * C_ + c;
  #pragma unroll
  for (int gy = 0; gy < 4; ++gy) {
    #pragma unroll
    for (int gx = 0; gx < 4; ++gx) {
      float v = 0.f;
      #pragma unroll
      for (int iy = 0; iy < 2; ++iy)
        #pragma unroll
        for (int ix = 0; ix < 2; ++ix)
          v += W[gy][iy] * W[gx][ix] * xv[iy][ix];
      int p = gy * 4 + gx;
      ob[(size_t)p * C_] = v * gm[p];
    }
  }
}

// ---------------------------------------------------------------------------
extern "C" void kernel_launch(void* const* d_in, const int* in_sizes, int n_in,
                              void* d_out, int out_size, void* d_ws, size_t ws_size,
                              hipStream_t stream) {
  const float* z   = (const float*)d_in[0];
  const float* x   = (const float*)d_in[1];
  const float* gth = (const float*)d_in[2];
  const float* gsg = (const float*)d_in[3];
  const float* glm = (const float*)d_in[4];
  const float* gps = (const float*)d_in[5];
  const float* ggm = (const float*)d_in[6];
  const float* gwt = (const float*)d_in[7];

  float* ws    = (float*)d_ws;
  float* zmax  = ws + WS_ZMAX;
  float* wsum  = ws + WS_WSUM;
  int*   idx   = (int*)(ws + WS_IDX);
  float* gmap1 = ws + WS_GMAP;
  float* out   = (float*)d_out;

  zmax_kernel  <<<B_, C_, 0, stream>>>(z, zmax);
  gmap_kernel  <<<1, 16, 0, stream>>>(gth, gsg, glm, gps, ggm, gwt, gmap1);
  winsum_kernel<<<B_ * 64, 256, 0, stream>>>(x, zmax, wsum);
  topk_kernel  <<<2, 32, 0, stream>>>(wsum, idx);
  gather_kernel<<<B_ * TOPK_, C_, 0, stream>>>(x, idx, gmap1, out);
}